// MyModel_87522843559019
// MI455X (gfx1250) — compile-verified
//
#include <hip/hip_runtime.h>

// Problem constants from the reference
#define B_ROWS   32
#define N_COLS   2048
#define OUT_COLS 100

typedef __attribute__((ext_vector_type(2))) float v2f;
typedef __attribute__((ext_vector_type(8))) float v8f;

// ---------------------------------------------------------------------------
// Kernel 1: collapse the [B,N,N] broadcast-add + reduce analytically.
//   S[b]      = sum_i (int)query[b,i]            (exact integer sum)
//   rem[b,j]  = (query[b,j] - float(N*c[b,j] + S[b]))^2
// One block per row b; 256 threads; LDS tree reduction for S[b].
// ---------------------------------------------------------------------------
__global__ void __launch_bounds__(256)
build_rem_kernel(const float* __restrict__ q, float* __restrict__ rem) {
  __shared__ int s_part[256];
  const int b = blockIdx.x;
  const int t = threadIdx.x;
  const float* row = q + b * N_COLS;

  int local = 0;
#pragma unroll
  for (int j = t; j < N_COLS; j += 256) local += (int)row[j];  // trunc-toward-0 == astype(int32)
  s_part[t] = local;
  __syncthreads();
  for (int s = 128; s > 0; s >>= 1) {
    if (t < s) s_part[t] += s_part[t + s];
    __syncthreads();
  }
  const int S = s_part[0];

#pragma unroll
  for (int j = t; j < N_COLS; j += 256) {
    float qv = row[j];
    int   cj = (int)qv;
    float fr = (float)(N_COLS * cj + S);   // < 2^24, exact in f32
    float d  = qv - fr;
    rem[b * N_COLS + j] = d * d;
  }
}

// ---------------------------------------------------------------------------
// Kernel 2: out[32,100] = rem[32,2048] @ W[2048,100] via V_WMMA_F32_16X16X4_F32.
// One wave32 per 16x16 output tile: 2 M-tiles x 7 N-tiles (100 padded to 112).
// K=2048 consumed 4 at a time -> 512 WMMA ops per tile.
//
// A-frag (16x4 f32, 2 VGPRs/lane, ISA 7.12.2):
//   lanes 0-15 : M=lane,    {K=k0, k0+1}
//   lanes 16-31: M=lane-16, {K=k0+2, k0+3}   -> one aligned float2 load from rem
// B-frag (4x16) mirrors with N on the lane axis; W is row-major [K][100] so two
// strided scalar loads. OOB columns (n>=100) are clamped + select-zeroed so
// EXEC stays all-1s (WMMA requirement).
// C/D layout: lanes 0-15 -> M=r (VGPR r), lanes 16-31 -> M=r+8; N = lane&15.
// ---------------------------------------------------------------------------
__global__ void __launch_bounds__(32)
wmma_gemm_kernel(const float* __restrict__ rem, const float* __restrict__ W,
                 float* __restrict__ out) {
  const int tile   = blockIdx.x;   // 0..13
  const int m_tile = tile / 7;     // 0..1
  const int n_tile = tile % 7;     // 0..6
  const int lane   = threadIdx.x;  // 0..31
  const int half   = lane >> 4;    // which K-pair this lane holds
  const int l16    = lane & 15;

  const int  m  = m_tile * 16 + l16;            // A row this lane loads
  const int  n  = n_tile * 16 + l16;            // B/D column this lane owns
  const bool nb = (n < OUT_COLS);
  const int  nc = nb ? n : (OUT_COLS - 1);      // clamped (in-bounds) load col
  const float nmask = nb ? 1.0f : 0.0f;

  const float* arow = rem + m * N_COLS + half * 2;

  v8f acc = {};
  for (int k0 = 0; k0 < N_COLS; k0 += 4) {
    const int ka = k0 + half * 2;
    v2f a = *(const v2f*)(arow + k0);           // 8B-aligned pair {K=ka, ka+1}
    v2f bf;
    bf.x = nmask * W[(ka + 0) * OUT_COLS + nc];
    bf.y = nmask * W[(ka + 1) * OUT_COLS + nc];
    acc = __builtin_amdgcn_wmma_f32_16x16x4_f32(
        /*neg_a=*/false, a, /*neg_b=*/false, bf,
        /*c_mod=*/(short)0, acc, /*reuse_a=*/false, /*reuse_b=*/false);
  }

  if (nb) {
#pragma unroll
    for (int r = 0; r < 8; ++r) {
      const int mm = m_tile * 16 + r + half * 8;
      out[mm * OUT_COLS + n] = acc[r];
    }
  }
}

// ---------------------------------------------------------------------------
extern "C" void kernel_launch(void* const* d_in, const int* in_sizes, int n_in,
                              void* d_out, int out_size, void* d_ws, size_t ws_size,
                              hipStream_t stream) {
  const float* q  = (const float*)d_in[0];   // [32, 2048]
  const float* W  = (const float*)d_in[1];   // [2048, 100]
  float*       out = (float*)d_out;          // [32, 100]
  float*       rem = (float*)d_ws;           // 32*2048 floats = 256 KB scratch

  build_rem_kernel<<<B_ROWS, 256, 0, stream>>>(q, rem);
  wmma_gemm_kernel<<<14, 32, 0, stream>>>(rem, W, out);
}